// IRadon_30331059044538
// MI455X (gfx1250) — compile-verified
//
#include <hip/hip_runtime.h>
#include <math.h>

// Problem constants (from reference): B=16, C=1, DET=512, T=180
#define DETN 512
#define NTH  180
#define NB   16

typedef __attribute__((ext_vector_type(2))) float v2f;
typedef __attribute__((ext_vector_type(8))) float v8f;

// ---------------------------------------------------------------------------
// Kernel 0: build the closed-form Ram-Lak spatial filter matrix H (512x512)
// and the cos/sin tables for 180 angles.
//   H[d,m] = 0.5                      if d==m
//          = -2/(pi^2 (d-m)^2)        if (d-m) odd
//          = 0                        otherwise
// (Derived exactly from the reference's FFT construction: f is even over the
//  1024-circle, so fourier = FFT(2f) and the filter is circular conv with 2f.)
// ---------------------------------------------------------------------------
__global__ void __launch_bounds__(256) fbp_init(float* __restrict__ H,
                                                float* __restrict__ trig) {
    int idx = blockIdx.x * 256 + threadIdx.x;
    if (idx < DETN * DETN) {
        int d = idx >> 9;
        int m = idx & (DETN - 1);
        int delta = d - m;
        float v;
        if (delta == 0) {
            v = 0.5f;
        } else if (delta & 1) {
            float fd = (float)delta;
            v = -2.0f / (9.869604401089358f * fd * fd);   // pi^2
        } else {
            v = 0.0f;
        }
        H[idx] = v;
    }
    if (idx < NTH) {
        float rad = (float)idx * 0.017453292519943295f;   // pi/180
        trig[idx]        = cosf(rad);
        trig[NTH + idx]  = sinf(rad);
    }
}

// ---------------------------------------------------------------------------
// Kernel 1: ramp filter as WMMA GEMM.
//   XF[d, th, b] = sum_k H[d,k] * x[b, k, th]
// M = d (512 -> 32 tiles), K = k (512 -> 128 steps of 4),
// N = th*16 + b (2880 -> 180 tiles; each N-tile = one theta, all 16 batches).
// One wave per 16x16 output tile; 128 x v_wmma_f32_16x16x4_f32 per wave.
// EXEC stays all-ones (no divergent control flow) as WMMA requires.
//
// ISA 16x4 f32 A layout: lanes 0-15 hold M=lane {VGPR0: K=0, VGPR1: K=1},
// lanes 16-31 hold M=lane-16 {VGPR0: K=2, VGPR1: K=3}. B (4x16) mirrored
// with N across lanes. D: VGPR j -> (M=j, N=lane) / (M=j+8, N=lane-16).
// ---------------------------------------------------------------------------
__global__ void __launch_bounds__(256) fbp_filter_gemm(
        const float* __restrict__ x,     // [B=16][DET=512][T=180]
        const float* __restrict__ H,     // [512][512]
        float* __restrict__ xf) {        // [DET=512][T=180][B=16]
    const int lane  = threadIdx.x & 31;
    const int wave  = threadIdx.x >> 5;
    const int tile  = blockIdx.x * 8 + wave;   // 720 blocks * 8 waves = 5760
    const int mt    = tile & 31;               // 0..31  (d-tile)
    const int th    = tile >> 5;               // 0..179 (theta = N-tile)
    const int d0    = mt << 4;

    const int row   = lane & 15;               // M (A) / b (B,N) within tile
    const int khalf = (lane >> 4) << 1;        // 0 for lanes 0-15, 2 for 16-31

    // A: H[d0+row][k+khalf .. k+khalf+1]  -> one b64 load per step
    const float* Aptr = H + (size_t)(d0 + row) * DETN + khalf;
    // B: x[b=row][k+khalf+{0,1}][th]      -> two b32 loads per step
    const float* Bptr = x + (size_t)row * (DETN * NTH) + (size_t)khalf * NTH + th;

    v8f c = {0.f, 0.f, 0.f, 0.f, 0.f, 0.f, 0.f, 0.f};

    for (int k = 0; k < DETN; k += 4) {
        v2f a = *(const v2f*)(Aptr + k);
        v2f b;
        b.x = Bptr[k * NTH];
        b.y = Bptr[k * NTH + NTH];
        c = __builtin_amdgcn_wmma_f32_16x16x4_f32(
                /*neg_a=*/false, a, /*neg_b=*/false, b,
                /*c_mod=*/(short)0, c, /*reuse_a=*/false, /*reuse_b=*/false);
    }

    // Store D: lanes 0-15 write rows d0+j (N=b=lane), lanes 16-31 rows d0+8+j.
    // Within a lane group the 16 b-values are contiguous -> 64B stores.
    const int b_out = lane & 15;
    const int mbase = d0 + ((lane >> 4) << 3);
    #pragma unroll
    for (int j = 0; j < 8; ++j) {
        int d = mbase + j;
        xf[((size_t)d * NTH + th) * NB + b_out] = c[j];
    }
}

// ---------------------------------------------------------------------------
// Kernel 2: backprojection. One thread = one pixel x one group of 4 batches.
// XF layout [det][theta][16 batches] makes each gather a coalesced float4.
// ---------------------------------------------------------------------------
__global__ void __launch_bounds__(256) fbp_backproj(
        const float* __restrict__ xf,    // [512][180][16]
        const float* __restrict__ trig,  // cos[180], sin[180]
        float* __restrict__ out) {       // [16][1][512][512]
    const int p  = blockIdx.x * 256 + threadIdx.x;  // 0..262143
    const int bg = blockIdx.y;                      // 0..3 (batch group of 4)
    const int px = p & (DETN - 1);
    const int py = p >> 9;

    const float ux = (2.0f * (float)px + 1.0f) * (1.0f / (float)DETN) - 1.0f;
    const float uy = (2.0f * (float)py + 1.0f) * (1.0f / (float)DETN) - 1.0f;

    float ax = 0.f, ay = 0.f, az = 0.f, aw = 0.f;

    for (int th = 0; th < NTH; ++th) {
        float ct = trig[th];
        float st = trig[NTH + th];
        float t  = ux * ct - uy * st;
        float iy = ((t + 1.0f) * (float)DETN - 1.0f) * 0.5f;
        float i0f = floorf(iy);
        float w1 = iy - i0f;
        float w0 = 1.0f - w1;
        int i0 = (int)i0f;
        int i1 = i0 + 1;
        float m0 = (i0 >= 0 && i0 < DETN) ? 1.0f : 0.0f;
        float m1 = (i1 >= 0 && i1 < DETN) ? 1.0f : 0.0f;
        int c0 = min(max(i0, 0), DETN - 1);
        int c1 = min(max(i1, 0), DETN - 1);

        const float4 g0 = *(const float4*)(xf + ((size_t)c0 * NTH + th) * NB + bg * 4);
        const float4 g1 = *(const float4*)(xf + ((size_t)c1 * NTH + th) * NB + bg * 4);

        float a0 = w0 * m0;
        float a1 = w1 * m1;
        ax = fmaf(a0, g0.x, fmaf(a1, g1.x, ax));
        ay = fmaf(a0, g0.y, fmaf(a1, g1.y, ay));
        az = fmaf(a0, g0.z, fmaf(a1, g1.z, az));
        aw = fmaf(a0, g0.w, fmaf(a1, g1.w, aw));
    }

    const float r2 = ux * ux + uy * uy;
    const float lim = 1.0f - 1.0f / (float)DETN;
    const float scale = (r2 <= lim) ? 0.008726646259971648f : 0.0f; // pi/360
    const size_t pix = (size_t)py * DETN + px;
    const size_t img = (size_t)DETN * DETN;
    out[(size_t)(bg * 4 + 0) * img + pix] = ax * scale;
    out[(size_t)(bg * 4 + 1) * img + pix] = ay * scale;
    out[(size_t)(bg * 4 + 2) * img + pix] = az * scale;
    out[(size_t)(bg * 4 + 3) * img + pix] = aw * scale;
}

// ---------------------------------------------------------------------------
extern "C" void kernel_launch(void* const* d_in, const int* in_sizes, int n_in,
                              void* d_out, int out_size, void* d_ws, size_t ws_size,
                              hipStream_t stream) {
    const float* x = (const float*)d_in[0];          // [16][1][512][180]
    float* H    = (float*)d_ws;                      // 512*512        = 262144 f
    float* xf   = H + DETN * DETN;                   // 512*180*16     = 1474560 f
    float* trig = xf + (size_t)DETN * NTH * NB;      // 2*180          = 360 f
    float* out  = (float*)d_out;                     // 16*512*512

    // Stage 0: filter matrix + trig tables (262144 threads).
    fbp_init<<<1024, 256, 0, stream>>>(H, trig);

    // Stage 1: ramp filter as WMMA GEMM. 5760 tiles / 8 waves-per-block.
    fbp_filter_gemm<<<720, 256, 0, stream>>>(x, H, xf);

    // Stage 2: backprojection. 262144 pixels x 4 batch groups.
    dim3 grid(1024, 4);
    fbp_backproj<<<grid, 256, 0, stream>>>(xf, trig, out);
}